// ConPrediction_1030792151123
// MI455X (gfx1250) — compile-verified
//
#include <hip/hip_runtime.h>
#include <math.h>
#include <stdint.h>

// ---------------- sizes from the reference ----------------
#define DIM      10
#define WIN      5
#define KSZ      11        // 2*WIN+1
#define L_GNN    3
#define L_CNN    3
#define L_NN     3
#define L_OUT    3
#define N_ATOMS  5000
#define N_SEQ    8000
#define L_PSSM   2000
#define N_RDKIT  200

typedef __attribute__((ext_vector_type(2))) float v2f;
typedef __attribute__((ext_vector_type(8))) float v8f;

// =====================================================================
// Gather: out[i][d] = emb[idx[i]][d]   (DIM columns)
// =====================================================================
__global__ void gather10_kernel(float* __restrict__ out,
                                const float* __restrict__ emb,
                                const int* __restrict__ idx, int n) {
    int t = blockIdx.x * blockDim.x + threadIdx.x;
    if (t >= n * DIM) return;
    int r = t / DIM, d = t - r * DIM;
    out[t] = emb[(long)idx[r] * DIM + d];
}

// =====================================================================
// GNN hs = relu(xs @ W^T + b), written zero-padded to 16 columns
// hsPad: [N_ATOMS][16]
// =====================================================================
__global__ void gnn_hs_kernel(float* __restrict__ hsPad,
                              const float* __restrict__ xs,
                              const float* __restrict__ W,
                              const float* __restrict__ b) {
    int t = blockIdx.x * blockDim.x + threadIdx.x;
    if (t >= N_ATOMS * 16) return;
    int row = t >> 4, col = t & 15;
    float v = 0.f;
    if (col < DIM) {
        float acc = b[col];
        const float* x = xs + row * DIM;
#pragma unroll
        for (int d = 0; d < DIM; ++d) acc += x[d] * W[col * DIM + d];
        v = fmaxf(acc, 0.f);
    }
    hsPad[t] = v;
}

// =====================================================================
// xs += adjacency @ hs   via V_WMMA_F32_16X16X4_F32
//   adjacency: [N_ATOMS][N_ATOMS] f32 row-major (the HBM-bound stream)
//   hsPad:     [N_ATOMS][16] f32 (cols 10..15 == 0)
// Block: 256 threads = 8 waves, covers 32 rows (two 16x16 M-tiles).
// K is processed in panels of KP staged into LDS with CDNA5 async
// global->LDS copies; 8 waves split the panel's k-blocks; each k-step
// feeds TWO wmma ops sharing one B fragment. Deterministic LDS reduce.
// =====================================================================
#define KP 512   // k-values per LDS panel (panel = KP*16 f32 = 32 KB)

__global__ __launch_bounds__(256) void gnn_spmm_wmma_kernel(
        float* __restrict__ xs,
        const float* __restrict__ adj,
        const float* __restrict__ hsPad) {
    __shared__ float panel[KP * 16];     // 32 KB B panel
    __shared__ float red[8][2][256];     // 16 KB partial C tiles

    const int tid   = threadIdx.x;
    const int wave  = tid >> 5;
    const int lane  = tid & 31;
    const int mloc  = lane & 15;         // M within tile (A), N (B)
    const int khalf = lane >> 4;         // 0 -> K{0,1}, 1 -> K{2,3}
    const int m0    = blockIdx.x * 32;

    // Clamp OOB rows to row 0 region end: junk only lands in never-stored
    // D rows, and keeps EXEC uniform (no divergence around the wmma).
    int r0 = m0 + mloc;       if (r0 >= N_ATOMS) r0 = N_ATOMS - 1;
    int r1 = m0 + 16 + mloc;  if (r1 >= N_ATOMS) r1 = N_ATOMS - 1;
    const float* __restrict__ a0row = adj + (size_t)r0 * N_ATOMS;
    const float* __restrict__ a1row = adj + (size_t)r1 * N_ATOMS;

    v8f c0 = {0.f, 0.f, 0.f, 0.f, 0.f, 0.f, 0.f, 0.f};
    v8f c1 = {0.f, 0.f, 0.f, 0.f, 0.f, 0.f, 0.f, 0.f};

    const uint32_t ldsbase = (uint32_t)(uintptr_t)(&panel[0]);

    for (int k0 = 0; k0 < N_ATOMS; k0 += KP) {
        const int kc    = (N_ATOMS - k0 < KP) ? (N_ATOMS - k0) : KP;
        const int bytes = kc * 64;       // kc rows x 16 f32

        // ---- async-stage B panel: global -> LDS, 16B per op ----
        {
            const char* src = (const char*)(hsPad + (size_t)k0 * 16);
            for (int off = tid * 16; off < bytes; off += 256 * 16) {
                uint32_t      laddr = ldsbase + (uint32_t)off;
                unsigned long long gaddr = (unsigned long long)(src + off);
                asm volatile("global_load_async_to_lds_b128 %0, %1, off"
                             :: "v"(laddr), "v"(gaddr) : "memory");
            }
            asm volatile("s_wait_asynccnt 0x0" ::: "memory");
        }
        // hint next A panel into cache while we compute this one
        if (k0 + KP < N_ATOMS) {
            __builtin_prefetch(a0row + k0 + KP, 0, 0);
            __builtin_prefetch(a1row + k0 + KP, 0, 0);
        }
        __syncthreads();

        // ---- compute: wave w takes k-blocks w, w+8, ... of this panel ----
        const int KBp = kc >> 2;         // k-blocks of 4 in panel
        for (int kb = wave; kb < KBp; kb += 8) {
            const int kk = kb * 4 + khalf * 2;   // panel-local k
            const int k  = k0 + kk;              // global k
            // A fragments (16x4 f32): vgpr0=K, vgpr1=K+1
            const float2 av0 = *(const float2*)(a0row + k);
            const float2 av1 = *(const float2*)(a1row + k);
            v2f a0; a0[0] = av0.x; a0[1] = av0.y;
            v2f a1; a1[0] = av1.x; a1[1] = av1.y;
            // B fragment (4x16 f32) from LDS: lane holds N=mloc
            const float* bp = panel + kk * 16 + mloc;
            v2f b; b[0] = bp[0]; b[1] = bp[16];
            c0 = __builtin_amdgcn_wmma_f32_16x16x4_f32(
                    false, a0, false, b, (short)0, c0, false, false);
            c1 = __builtin_amdgcn_wmma_f32_16x16x4_f32(
                    false, a1, false, b, (short)0, c1, false, false);
        }
        __syncthreads();                 // panel reuse barrier
    }

    // ---- deterministic cross-wave reduction of the two C tiles ----
    // C/D layout: vgpr j, lanes0-15 -> (M=j, N=lane); lanes16-31 -> (M=j+8, N=lane-16)
#pragma unroll
    for (int j = 0; j < 8; ++j) {
        red[wave][0][(j + khalf * 8) * 16 + mloc] = c0[j];
        red[wave][1][(j + khalf * 8) * 16 + mloc] = c1[j];
    }
    __syncthreads();

    const int M = tid >> 4, N = tid & 15;
#pragma unroll
    for (int t = 0; t < 2; ++t) {
        float s = 0.f;
#pragma unroll
        for (int w = 0; w < 8; ++w) s += red[w][t][tid];
        const int r = m0 + t * 16 + M;
        if (r < N_ATOMS && N < DIM) xs[r * DIM + N] += s;
    }
}

// =====================================================================
// 11x11 same-pad conv + relu over an [H][10] image (1 in / 1 out channel)
// =====================================================================
__global__ void conv11_kernel(float* __restrict__ out,
                              const float* __restrict__ in,
                              const float* __restrict__ K,
                              const float* __restrict__ bias, int H) {
    int t = blockIdx.x * blockDim.x + threadIdx.x;
    if (t >= H * DIM) return;
    int h = t / DIM, w = t - h * DIM;
    float acc = bias[0];
#pragma unroll
    for (int kh = 0; kh < KSZ; ++kh) {
        int ih = h + kh - WIN;
        if ((unsigned)ih >= (unsigned)H) continue;
        const float* irow = in + ih * DIM;
        const float* krow = K + kh * KSZ;
#pragma unroll
        for (int kw = 0; kw < KSZ; ++kw) {
            int iw = w + kw - WIN;
            if ((unsigned)iw < (unsigned)DIM) acc += krow[kw] * irow[iw];
        }
    }
    out[t] = fmaxf(acc, 0.f);
}

// =====================================================================
// Row-wise out = relu(in @ W^T + b), in/out [n][10]
// =====================================================================
__global__ void lin_rows_relu_kernel(float* __restrict__ out,
                                     const float* __restrict__ in,
                                     const float* __restrict__ W,
                                     const float* __restrict__ b, int n) {
    int t = blockIdx.x * blockDim.x + threadIdx.x;
    if (t >= n * DIM) return;
    int r = t / DIM, col = t - r * DIM;
    float acc = b[col];
    const float* x = in + r * DIM;
#pragma unroll
    for (int d = 0; d < DIM; ++d) acc += x[d] * W[col * DIM + d];
    out[t] = fmaxf(acc, 0.f);
}

// =====================================================================
// out[d] = mean over rows of in[*][d]; single block -> deterministic
// =====================================================================
__global__ __launch_bounds__(256) void mean_rows_kernel(
        float* __restrict__ out, const float* __restrict__ in, int n) {
    __shared__ float part[256][DIM];
    float acc[DIM];
#pragma unroll
    for (int d = 0; d < DIM; ++d) acc[d] = 0.f;
    for (int r = threadIdx.x; r < n; r += 256) {
        const float* row = in + r * DIM;
#pragma unroll
        for (int d = 0; d < DIM; ++d) acc[d] += row[d];
    }
#pragma unroll
    for (int d = 0; d < DIM; ++d) part[threadIdx.x][d] = acc[d];
    __syncthreads();
    if (threadIdx.x < DIM) {
        float s = 0.f;
        for (int t = 0; t < 256; ++t) s += part[t][threadIdx.x];
        out[threadIdx.x] = s / (float)n;
    }
}

// =====================================================================
// q_out = relu(q @ Wa^T + b), q is [1][10]
// =====================================================================
__global__ void lin_vec_relu_kernel(float* __restrict__ out,
                                    const float* __restrict__ q,
                                    const float* __restrict__ W,
                                    const float* __restrict__ b) {
    int c = threadIdx.x;
    if (c >= DIM) return;
    float acc = b[c];
#pragma unroll
    for (int d = 0; d < DIM; ++d) acc += q[d] * W[c * DIM + d];
    out[c] = fmaxf(acc, 0.f);
}

// =====================================================================
// out[d] = (1/n) * sum_i tanh(h . hs_i) * hs_i[d]; single block
// =====================================================================
__global__ __launch_bounds__(256) void attn_mean_kernel(
        float* __restrict__ out, const float* __restrict__ h,
        const float* __restrict__ hs, int n) {
    __shared__ float part[256][DIM];
    __shared__ float hq[DIM];
    if (threadIdx.x < DIM) hq[threadIdx.x] = h[threadIdx.x];
    __syncthreads();
    float acc[DIM];
#pragma unroll
    for (int d = 0; d < DIM; ++d) acc[d] = 0.f;
    for (int r = threadIdx.x; r < n; r += 256) {
        const float* row = hs + r * DIM;
        float dot = 0.f;
#pragma unroll
        for (int d = 0; d < DIM; ++d) dot += hq[d] * row[d];
        float w = tanhf(dot);
#pragma unroll
        for (int d = 0; d < DIM; ++d) acc[d] += w * row[d];
    }
#pragma unroll
    for (int d = 0; d < DIM; ++d) part[threadIdx.x][d] = acc[d];
    __syncthreads();
    if (threadIdx.x < DIM) {
        float s = 0.f;
        for (int t = 0; t < 256; ++t) s += part[t][threadIdx.x];
        out[threadIdx.x] = s / (float)n;
    }
}

// =====================================================================
// Tail: pairwise outer-product module + concat + output MLP (tiny, scalar)
// =====================================================================
__global__ void tail_kernel(float* __restrict__ out,
                            const float* __restrict__ sub,
                            const float* __restrict__ prot,
                            const float* __restrict__ Wp_w, const float* __restrict__ Wp_b,
                            const float* __restrict__ Po_w, const float* __restrict__ Po_b,
                            const float* __restrict__ Wo_w, const float* __restrict__ Wo_b,
                            const float* __restrict__ Wi_w, const float* __restrict__ Wi_b) {
    if (threadIdx.x != 0 || blockIdx.x != 0) return;
    float pcf[DIM], ppf[DIM];
    for (int c = 0; c < DIM; ++c) {
        float a = Wp_b[c], b = Wp_b[c];
        for (int d = 0; d < DIM; ++d) {
            a += sub[d]  * Wp_w[c * DIM + d];
            b += prot[d] * Wp_w[c * DIM + d];
        }
        pcf[c] = (a > 0.f) ? a : 0.01f * a;   // leaky_relu slope 0.01
        ppf[c] = (b > 0.f) ? b : 0.01f * b;
    }
    float hid[DIM][DIM], tmp[DIM][DIM];
    for (int r = 0; r < DIM; ++r)
        for (int c = 0; c < DIM; ++c)
            hid[r][c] = 1.f / (1.f + expf(-(ppf[r] * pcf[c])));
    for (int l = 0; l < L_NN; ++l) {
        for (int r = 0; r < DIM; ++r)
            for (int c = 0; c < DIM; ++c) {
                float acc = Po_b[l * DIM + c];
                for (int d = 0; d < DIM; ++d)
                    acc += hid[r][d] * Po_w[l * DIM * DIM + c * DIM + d];
                tmp[r][c] = fmaxf(acc, 0.f);
            }
        for (int r = 0; r < DIM; ++r)
            for (int c = 0; c < DIM; ++c) hid[r][c] = tmp[r][c];
    }
    float pair[DIM];
    for (int c = 0; c < DIM; ++c) pair[c] = 0.f;
    for (int r = 0; r < DIM; ++r)
        for (int c = 0; c < DIM; ++c) {
            float acc = Wp_b[c];
            for (int d = 0; d < DIM; ++d) acc += hid[r][d] * Wp_w[c * DIM + d];
            pair[c] += fmaxf(acc, 0.f);
        }
    for (int c = 0; c < DIM; ++c) pair[c] *= (1.f / (float)DIM);

    float cat[3 * DIM], cat2[3 * DIM];
    for (int d = 0; d < DIM; ++d) {
        cat[d] = sub[d]; cat[DIM + d] = prot[d]; cat[2 * DIM + d] = pair[d];
    }
    for (int j = 0; j < L_OUT; ++j) {
        for (int c = 0; c < 3 * DIM; ++c) {
            float acc = Wo_b[j * 3 * DIM + c];
            for (int d = 0; d < 3 * DIM; ++d)
                acc += cat[d] * Wo_w[j * 9 * DIM * DIM + c * 3 * DIM + d];
            cat2[c] = fmaxf(acc, 0.f);
        }
        for (int c = 0; c < 3 * DIM; ++c) cat[c] = cat2[c];
    }
    float o = Wi_b[0];
    for (int d = 0; d < 3 * DIM; ++d) o += cat[d] * Wi_w[d];
    out[0] = o;
}

// =====================================================================
// Host-side orchestration
// =====================================================================
static inline int cdiv(int a, int b) { return (a + b - 1) / b; }

extern "C" void kernel_launch(void* const* d_in, const int* in_sizes, int n_in,
                              void* d_out, int out_size, void* d_ws, size_t ws_size,
                              hipStream_t stream) {
    (void)in_sizes; (void)n_in; (void)out_size; (void)ws_size;
    const int*   fingerprints = (const int*)  d_in[0];
    const float* adjacency    = (const float*)d_in[1];
    const int*   words        = (const int*)  d_in[2];
    const float* pssms        = (const float*)d_in[3];
    const float* rdkitfeat    = (const float*)d_in[4];
    const float* emb_fp       = (const float*)d_in[5];
    const float* emb_word     = (const float*)d_in[6];
    const float* Wg_w = (const float*)d_in[7];
    const float* Wg_b = (const float*)d_in[8];
    const float* Wc_k = (const float*)d_in[9];
    const float* Wc_b = (const float*)d_in[10];
    const float* Wn_w = (const float*)d_in[11];
    const float* Wn_b = (const float*)d_in[12];
    const float* Wa_w = (const float*)d_in[13];
    const float* Wa_b = (const float*)d_in[14];
    const float* Wp_w = (const float*)d_in[15];
    const float* Wp_b = (const float*)d_in[16];
    const float* Po_w = (const float*)d_in[17];
    const float* Po_b = (const float*)d_in[18];
    const float* Wo_w = (const float*)d_in[19];
    const float* Wo_b = (const float*)d_in[20];
    const float* Wi_w = (const float*)d_in[21];
    const float* Wi_b = (const float*)d_in[22];
    float* out = (float*)d_out;

    // workspace layout (floats)
    float* ws        = (float*)d_ws;
    float* xs        = ws;                 // 50000   [5000][10]
    float* hsPad     = ws + 50000;         // 80000   [5000][16]
    float* bufA      = ws + 130000;        // 80000   [<=8000][10]
    float* bufB      = ws + 210000;        // 80000
    float* hsAtt     = ws + 290000;        // 80000
    float* rdkit_vec = ws + 370000;        // 16
    float* pssm_vec  = ws + 370016;        // 16
    float* q_h       = ws + 370032;        // 16
    float* subst_vec = ws + 370048;        // 16
    float* prot_vec  = ws + 370064;        // 16

    const int B = 256;

    // ---- GNN: xs = emb_fp[fingerprints]; 3x (hs=relu(lin); xs += A@hs) ----
    gather10_kernel<<<cdiv(N_ATOMS * DIM, B), B, 0, stream>>>(xs, emb_fp, fingerprints, N_ATOMS);
    const int spmmBlocks = cdiv(N_ATOMS, 32);   // 157 (32 rows per block)
    for (int l = 0; l < L_GNN; ++l) {
        gnn_hs_kernel<<<cdiv(N_ATOMS * 16, B), B, 0, stream>>>(
            hsPad, xs, Wg_w + l * DIM * DIM, Wg_b + l * DIM);
        gnn_spmm_wmma_kernel<<<spmmBlocks, B, 0, stream>>>(xs, adjacency, hsPad);
    }

    // ---- rdkit MLP + mean of relu(lin(., Wa)) ----
    lin_rows_relu_kernel<<<cdiv(N_RDKIT * DIM, B), B, 0, stream>>>(bufA, rdkitfeat, Wn_w, Wn_b, N_RDKIT);
    lin_rows_relu_kernel<<<cdiv(N_RDKIT * DIM, B), B, 0, stream>>>(bufB, bufA, Wn_w + 100, Wn_b + 10, N_RDKIT);
    lin_rows_relu_kernel<<<cdiv(N_RDKIT * DIM, B), B, 0, stream>>>(bufA, bufB, Wn_w + 200, Wn_b + 20, N_RDKIT);
    lin_rows_relu_kernel<<<cdiv(N_RDKIT * DIM, B), B, 0, stream>>>(bufB, bufA, Wa_w, Wa_b, N_RDKIT);
    mean_rows_kernel<<<1, B, 0, stream>>>(rdkit_vec, bufB, N_RDKIT);

    // ---- substrate = attention(rdkit_vec, conv_stack(compound)) ----
    conv11_kernel<<<cdiv(N_ATOMS * DIM, B), B, 0, stream>>>(bufA, xs,   Wc_k,       Wc_b,     N_ATOMS);
    conv11_kernel<<<cdiv(N_ATOMS * DIM, B), B, 0, stream>>>(bufB, bufA, Wc_k + 121, Wc_b + 1, N_ATOMS);
    conv11_kernel<<<cdiv(N_ATOMS * DIM, B), B, 0, stream>>>(bufA, bufB, Wc_k + 242, Wc_b + 2, N_ATOMS);
    lin_rows_relu_kernel<<<cdiv(N_ATOMS * DIM, B), B, 0, stream>>>(hsAtt, bufA, Wa_w, Wa_b, N_ATOMS);
    lin_vec_relu_kernel<<<1, 32, 0, stream>>>(q_h, rdkit_vec, Wa_w, Wa_b);
    attn_mean_kernel<<<1, B, 0, stream>>>(subst_vec, q_h, hsAtt, N_ATOMS);

    // ---- pssm vector: conv stack + lin + mean ----
    conv11_kernel<<<cdiv(L_PSSM * DIM, B), B, 0, stream>>>(bufA, pssms, Wc_k,       Wc_b,     L_PSSM);
    conv11_kernel<<<cdiv(L_PSSM * DIM, B), B, 0, stream>>>(bufB, bufA,  Wc_k + 121, Wc_b + 1, L_PSSM);
    conv11_kernel<<<cdiv(L_PSSM * DIM, B), B, 0, stream>>>(bufA, bufB,  Wc_k + 242, Wc_b + 2, L_PSSM);
    lin_rows_relu_kernel<<<cdiv(L_PSSM * DIM, B), B, 0, stream>>>(bufB, bufA, Wa_w, Wa_b, L_PSSM);
    mean_rows_kernel<<<1, B, 0, stream>>>(pssm_vec, bufB, L_PSSM);

    // ---- protein = attention(pssm_vec, conv_stack(emb_word[words])) ----
    gather10_kernel<<<cdiv(N_SEQ * DIM, B), B, 0, stream>>>(bufA, emb_word, words, N_SEQ);
    conv11_kernel<<<cdiv(N_SEQ * DIM, B), B, 0, stream>>>(bufB, bufA, Wc_k,       Wc_b,     N_SEQ);
    conv11_kernel<<<cdiv(N_SEQ * DIM, B), B, 0, stream>>>(bufA, bufB, Wc_k + 121, Wc_b + 1, N_SEQ);
    conv11_kernel<<<cdiv(N_SEQ * DIM, B), B, 0, stream>>>(bufB, bufA, Wc_k + 242, Wc_b + 2, N_SEQ);
    lin_rows_relu_kernel<<<cdiv(N_SEQ * DIM, B), B, 0, stream>>>(hsAtt, bufB, Wa_w, Wa_b, N_SEQ);
    lin_vec_relu_kernel<<<1, 32, 0, stream>>>(q_h, pssm_vec, Wa_w, Wa_b);
    attn_mean_kernel<<<1, B, 0, stream>>>(prot_vec, q_h, hsAtt, N_SEQ);

    // ---- pairwise module + output MLP ----
    tail_kernel<<<1, 32, 0, stream>>>(out, subst_vec, prot_vec,
                                      Wp_w, Wp_b, Po_w, Po_b, Wo_w, Wo_b, Wi_w, Wi_b);
}